// CliffordSpectralConv2d_73005854098017
// MI455X (gfx1250) — compile-verified
//
#include <hip/hip_runtime.h>

// ---------------------------------------------------------------------------
// CliffordSpectralConv2d on gfx1250 via partial-DFT-as-GEMM with
// V_WMMA_F32_16X16X4_F32.  B=4, Cin=Cout=32, H=W=256, m1=m2=32.
// ---------------------------------------------------------------------------

typedef __attribute__((ext_vector_type(2))) float v2f;
typedef __attribute__((ext_vector_type(8))) float v8f;

__device__ __forceinline__ v8f wmma4(v2f a, v2f b, v8f c) {
    // D(16x16) += A(16x4) * B(4x16), all f32, wave32
    return __builtin_amdgcn_wmma_f32_16x16x4_f32(
        /*neg_a=*/false, a, /*neg_b=*/false, b,
        /*c_mod=*/(short)0, c, /*reuse_a=*/false, /*reuse_b=*/false);
}

__device__ __forceinline__ v2f ld2(const float* p) { return *(const v2f*)p; }

// ---- workspace layout (float offsets) -------------------------------------
#define OFF_TF_CRE   0u          // fwd twiddle cos            [64][256]
#define OFF_TF_CIM   16384u      // fwd twiddle -sin           [64][256]
#define OFF_TF_CIMN  32768u      // fwd twiddle +sin           [64][256]
#define OFF_TI_RE    49152u      // inv twiddle cos/256        [256][64]
#define OFF_TI_IM    65536u      // inv twiddle sin/256        [256][64]
#define OFF_TI_IMN   81920u      // inv twiddle -sin/256       [256][64]
#define OFF_A1       98304u      // stage-A out / inv-stage-1 out (reused)
                                 // [p][part][128 bc][256 h][64 kw] = 8388608
#define OFF_INP      84870 ## 12u
#undef OFF_INP
#define OFF_INP      (OFF_A1 + 8388608u)   // einsum input  [4 b][128 i][4096 mode]
#define OFF_MIX      (OFF_INP + 2097152u)  // einsum output [4 b][128 o4][4096 mode]
#define OFF_WT       (OFF_MIX + 2097152u)  // wT [4 cmp][4096 mode][32 o][32 i]

#define A1_PLANE     2097152u    // 128*256*64

// Clifford kernel structure: kernel[a*32+o, bb*32+i] = SGN[a][bb]*w[CMP[a][bb],o,i]
__constant__ int   d_CMP[16] = {0,1,2,3, 1,0,3,2, 2,3,0,1, 3,2,1,0};
__constant__ float d_SGN[16] = {1.f,1.f,1.f,-1.f, 1.f,1.f,-1.f,1.f,
                                1.f,1.f,1.f,-1.f, 1.f,1.f,-1.f,1.f};

// ---------------------------------------------------------------------------
// 0) twiddle tables.  mode c -> frequency kc = c<32 ? c : c+192
// ---------------------------------------------------------------------------
__global__ void k_init_tables(float* __restrict__ ws) {
    int t = blockIdx.x * 256 + threadIdx.x;       // 16384 threads
    int c = t >> 8, n = t & 255;
    int kc = (c < 32) ? c : (c + 192);
    int ph = (kc * n) & 255;                      // exact phase mod 256
    float th = (float)ph * 0.024543692602170259f; // 2*pi/256
    float s, co;
    __sincosf(th, &s, &co);
    ws[OFF_TF_CRE  + c * 256 + n] = co;           // e^{-i th}
    ws[OFF_TF_CIM  + c * 256 + n] = -s;
    ws[OFF_TF_CIMN + c * 256 + n] = s;
    const float inv = 1.0f / 256.0f;              // e^{+i th}/256
    ws[OFF_TI_RE  + n * 64 + c] = co * inv;
    ws[OFF_TI_IM  + n * 64 + c] = s * inv;
    ws[OFF_TI_IMN + n * 64 + c] = -s * inv;
}

// ---------------------------------------------------------------------------
// 1) weight transpose: w[cmp][o][i][x][y] -> wT[cmp][mode][o][i]
//    per-cmp 1024 x 4096 transpose, LDS tiled.
// ---------------------------------------------------------------------------
__global__ void k_transpose_w(const float* __restrict__ w, float* __restrict__ ws) {
    __shared__ float tile[32][33];
    float* wT = ws + OFF_WT;
    int cmp = blockIdx.z;
    int tx = threadIdx.x, ty = threadIdx.y;
    int col = blockIdx.x * 32 + tx;               // mode
    int row = blockIdx.y * 32 + ty;               // (o,i)
    for (int j = 0; j < 32; j += 8)
        tile[ty + j][tx] = w[((size_t)cmp * 1024 + row + j) * 4096 + col];
    __syncthreads();
    int orow = blockIdx.y * 32 + tx;              // (o,i)
    int ocol = blockIdx.x * 32 + ty;              // mode
    for (int j = 0; j < 32; j += 8)
        wT[((size_t)cmp * 4096 + ocol + j) * 1024 + orow] = tile[tx][ty + j];
}

// ---------------------------------------------------------------------------
// 2) forward DFT along W (all rows, 64 kept modes).  M=(b,c,h)=32768, N=64, K=256
//    d1 = x0 + i*x3, d2 = x1 + i*x2.  8 WMMA per K-step (2 duals x complex).
// ---------------------------------------------------------------------------
__global__ __launch_bounds__(128) void k_dft_w(const float* __restrict__ x,
                                               float* __restrict__ ws) {
    const int lane = threadIdx.x & 31, half = lane >> 4, lid = lane & 15;
    int gt = blockIdx.x * 4 + (threadIdx.x >> 5);   // 8192 wave-tiles
    int mt = gt >> 2, nt = gt & 3;
    int row0 = mt * 16, kw0 = nt * 16;
    int b = row0 >> 13, rem = row0 & 8191, ch = rem >> 8, h0 = rem & 255;
    int bc = b * 32 + ch;

    const float4* xb = (const float4*)x + ((size_t)bc * 256 + h0 + lid) * 256;
    const float* cre  = ws + OFF_TF_CRE  + (kw0 + lid) * 256;
    const float* cim  = ws + OFF_TF_CIM  + (kw0 + lid) * 256;
    const float* cimn = ws + OFF_TF_CIMN + (kw0 + lid) * 256;

    v8f a1r = {}, a1i = {}, a2r = {}, a2i = {};
    for (int k0 = 0; k0 < 256; k0 += 4) {
        int w0 = k0 + 2 * half;
        float4 f0 = xb[w0], f1 = xb[w0 + 1];
        v2f d1re{f0.x, f1.x}, d1im{f0.w, f1.w};
        v2f d2re{f0.y, f1.y}, d2im{f0.z, f1.z};
        v2f tre = ld2(cre + w0), tim = ld2(cim + w0), tmn = ld2(cimn + w0);
        a1r = wmma4(d1re, tre, a1r); a1r = wmma4(d1im, tmn, a1r);
        a1i = wmma4(d1re, tim, a1i); a1i = wmma4(d1im, tre, a1i);
        a2r = wmma4(d2re, tre, a2r); a2r = wmma4(d2im, tmn, a2r);
        a2i = wmma4(d2re, tim, a2i); a2i = wmma4(d2im, tre, a2i);
    }
    float* A1 = ws + OFF_A1;
    int kw = kw0 + lid;
    for (int v = 0; v < 8; v++) {
        int h = h0 + v + 8 * half;
        size_t base = (size_t)bc * 16384 + h * 64 + kw;
        A1[0 * A1_PLANE + base] = a1r[v];   // p0 re
        A1[1 * A1_PLANE + base] = a1i[v];   // p0 im
        A1[2 * A1_PLANE + base] = a2r[v];   // p1 re
        A1[3 * A1_PLANE + base] = a2i[v];   // p1 im
    }
}

// ---------------------------------------------------------------------------
// 3) forward DFT along H + blade concat + corner crop.
//    D[kh][col], col=(p,b,ch,kw): M=64, N=16384, K=256.  A = twiddles, B = A1.
// ---------------------------------------------------------------------------
__global__ __launch_bounds__(128) void k_dft_h(float* __restrict__ ws) {
    const int lane = threadIdx.x & 31, half = lane >> 4, lid = lane & 15;
    int gt = blockIdx.x * 4 + (threadIdx.x >> 5);   // 4096 wave-tiles
    int mt = gt & 3, nt = gt >> 2;
    int kh0 = mt * 16, col0 = nt * 16;
    int kw0 = col0 & 63, pbch = col0 >> 6;
    int ch = pbch & 31, bb = (pbch >> 5) & 3, p = pbch >> 7;
    int bc = bb * 32 + ch;

    const float* A1  = ws + OFF_A1;
    const float* Are = A1 + ((size_t)(p * 2 + 0) * 128 + bc) * 16384 + kw0 + lid;
    const float* Aim = A1 + ((size_t)(p * 2 + 1) * 128 + bc) * 16384 + kw0 + lid;
    const float* cre  = ws + OFF_TF_CRE  + (kh0 + lid) * 256;
    const float* cim  = ws + OFF_TF_CIM  + (kh0 + lid) * 256;
    const float* cimn = ws + OFF_TF_CIMN + (kh0 + lid) * 256;

    v8f ar = {}, ai = {};
    for (int k0 = 0; k0 < 256; k0 += 4) {
        int h = k0 + 2 * half;
        v2f tre = ld2(cre + h), tim = ld2(cim + h), tmn = ld2(cimn + h);
        v2f bre{Are[(size_t)h * 64], Are[(size_t)(h + 1) * 64]};
        v2f bim{Aim[(size_t)h * 64], Aim[(size_t)(h + 1) * 64]};
        ar = wmma4(tre, bre, ar); ar = wmma4(tmn, bim, ar);
        ai = wmma4(tim, bre, ai); ai = wmma4(tre, bim, ai);
    }
    float* inp = ws + OFF_INP;
    int jre = (p == 0) ? 0 : 1;   // d1f.re -> block0, d2f.re -> block1
    int jim = (p == 0) ? 3 : 2;   // d1f.im -> block3, d2f.im -> block2
    int kw = kw0 + lid;
    for (int v = 0; v < 8; v++) {
        int kh = kh0 + v + 8 * half;
        int mode = kh * 64 + kw;
        inp[((size_t)bb * 128 + jre * 32 + ch) * 4096 + mode] = ar[v];
        inp[((size_t)bb * 128 + jim * 32 + ch) * 4096 + mode] = ai[v];
    }
}

// ---------------------------------------------------------------------------
// 4) per-mode Clifford einsum: out[b][o4] = sum_i inp[b][i] * K[o4][i].
//    M=4 (padded to 16), N=128, K=128, batched over 4096 modes.
// ---------------------------------------------------------------------------
__global__ __launch_bounds__(128) void k_modemix(float* __restrict__ ws) {
    const int lane = threadIdx.x & 31, half = lane >> 4, lid = lane & 15;
    int gt = blockIdx.x * 4 + (threadIdx.x >> 5);   // 32768 wave-tiles
    int nt = gt & 7, mode = gt >> 3;
    int o4_0 = nt * 16;
    int ablk = o4_0 >> 5;
    int o = (o4_0 & 31) + lid;                      // o within [0,32)

    const float* inp = ws + OFF_INP;
    const float* wT  = ws + OFF_WT;
    float* mix = ws + OFF_MIX;
    const float* ainp = inp + (size_t)(lid & 3) * 524288 + mode;  // rows 4..15 dup row

    v8f acc = {};
    for (int k0 = 0; k0 < 128; k0 += 4) {
        int bblk = k0 >> 5;
        int cmp  = d_CMP[ablk * 4 + bblk];
        float sg = d_SGN[ablk * 4 + bblk];
        int kk = k0 + 2 * half;
        v2f a{ainp[(size_t)kk * 4096], ainp[(size_t)(kk + 1) * 4096]};
        v2f bf = ld2(wT + (((size_t)cmp * 4096 + mode) * 32 + o) * 32 + (kk & 31));
        bf = bf * sg;
        acc = wmma4(a, bf, acc);
    }
    if (half == 0) {
        for (int v = 0; v < 4; v++)     // only b-rows 0..3 are real
            mix[((size_t)v * 128 + o4_0 + lid) * 4096 + mode] = acc[v];
    }
}

// ---------------------------------------------------------------------------
// 5) inverse DFT along H: E[h][col], col=(b,o,kw): M=256, N=8192/p, K=64.
//    spec: p0 -> (blade0, blade3), p1 -> (blade1, blade2).  Writes over A1.
// ---------------------------------------------------------------------------
__global__ __launch_bounds__(128) void k_idft_h(float* __restrict__ ws) {
    const int lane = threadIdx.x & 31, half = lane >> 4, lid = lane & 15;
    int gt = blockIdx.x * 4 + (threadIdx.x >> 5);   // 16384 wave-tiles
    int p = gt >> 13, r = gt & 8191;
    int mt = r & 15, nt = r >> 4;
    int h0 = mt * 16, col0 = nt * 16;
    int kw0 = col0 & 63, bo = col0 >> 6, o = bo & 31, b = bo >> 5;
    int are = (p == 0) ? 0 : 1, aim = (p == 0) ? 3 : 2;

    const float* mix = ws + OFF_MIX;
    const float* Sre = mix + ((size_t)b * 128 + are * 32 + o) * 4096 + kw0 + lid;
    const float* Sim = mix + ((size_t)b * 128 + aim * 32 + o) * 4096 + kw0 + lid;
    const float* ire = ws + OFF_TI_RE  + (h0 + lid) * 64;
    const float* iim = ws + OFF_TI_IM  + (h0 + lid) * 64;
    const float* imn = ws + OFF_TI_IMN + (h0 + lid) * 64;

    v8f er = {}, ei = {};
    for (int k0 = 0; k0 < 64; k0 += 4) {
        int c = k0 + 2 * half;
        v2f tre = ld2(ire + c), tim = ld2(iim + c), tmn = ld2(imn + c);
        v2f bre{Sre[(size_t)c * 64], Sre[(size_t)(c + 1) * 64]};
        v2f bim{Sim[(size_t)c * 64], Sim[(size_t)(c + 1) * 64]};
        er = wmma4(tre, bre, er); er = wmma4(tmn, bim, er);
        ei = wmma4(tim, bre, ei); ei = wmma4(tre, bim, ei);
    }
    float* E = ws + OFF_A1;     // A1 storage reused
    int kw = kw0 + lid;
    size_t bre_ = ((size_t)(p * 2 + 0) * 128 + b * 32 + o) * 16384;
    size_t bim_ = ((size_t)(p * 2 + 1) * 128 + b * 32 + o) * 16384;
    for (int v = 0; v < 8; v++) {
        int h = h0 + v + 8 * half;
        E[bre_ + h * 64 + kw] = er[v];
        E[bim_ + h * 64 + kw] = ei[v];
    }
}

// ---------------------------------------------------------------------------
// 6) inverse DFT along W + recombine blades into final (B,O,H,W,4) output.
//    D[w][col], col=(b,o,h): M=256, N=32768/p, K=64.
// ---------------------------------------------------------------------------
__global__ __launch_bounds__(128) void k_idft_w(const float* __restrict__ ws,
                                                float* __restrict__ out) {
    const int lane = threadIdx.x & 31, half = lane >> 4, lid = lane & 15;
    int gt = blockIdx.x * 4 + (threadIdx.x >> 5);   // 65536 wave-tiles
    int p = gt >> 15, r = gt & 32767;
    int mt = r & 15, nt = r >> 4;
    int w0 = mt * 16, col0 = nt * 16;
    int hc = col0 & 255, bo = col0 >> 8, o = bo & 31, b = bo >> 5;

    const float* E = ws + OFF_A1;
    const float* Ere = E + (((size_t)(p * 2 + 0) * 128 + b * 32 + o) * 256 + hc + lid) * 64;
    const float* Eim = E + (((size_t)(p * 2 + 1) * 128 + b * 32 + o) * 256 + hc + lid) * 64;
    const float* ire = ws + OFF_TI_RE  + (w0 + lid) * 64;
    const float* iim = ws + OFF_TI_IM  + (w0 + lid) * 64;
    const float* imn = ws + OFF_TI_IMN + (w0 + lid) * 64;

    v8f qr = {}, qi = {};
    for (int k0 = 0; k0 < 64; k0 += 4) {
        int c = k0 + 2 * half;
        v2f tre = ld2(ire + c), tim = ld2(iim + c), tmn = ld2(imn + c);
        v2f bre = ld2(Ere + c);
        v2f bim = ld2(Eim + c);
        qr = wmma4(tre, bre, qr); qr = wmma4(tmn, bim, qr);
        qi = wmma4(tim, bre, qi); qi = wmma4(tre, bim, qi);
    }
    int cre = (p == 0) ? 0 : 1;   // o1.re->comp0, o2.re->comp1
    int cim = (p == 0) ? 3 : 2;   // o1.im->comp3, o2.im->comp2
    int h = hc + lid;
    size_t rowbase = (((size_t)b * 32 + o) * 256 + h) * 1024;
    for (int v = 0; v < 8; v++) {
        int w = w0 + v + 8 * half;
        out[rowbase + w * 4 + cre] = qr[v];
        out[rowbase + w * 4 + cim] = qi[v];
    }
}

// ---------------------------------------------------------------------------
extern "C" void kernel_launch(void* const* d_in, const int* in_sizes, int n_in,
                              void* d_out, int out_size, void* d_ws, size_t ws_size,
                              hipStream_t stream) {
    (void)in_sizes; (void)n_in; (void)out_size; (void)ws_size;
    const float* x = (const float*)d_in[0];       // (4,32,256,256,4)
    const float* w = (const float*)d_in[1];       // (4,32,32,64,64)
    float* out = (float*)d_out;                   // (4,32,256,256,4)
    float* ws  = (float*)d_ws;

    k_init_tables<<<64, 256, 0, stream>>>(ws);
    k_transpose_w<<<dim3(128, 32, 4), dim3(32, 8), 0, stream>>>(w, ws);
    k_dft_w   <<<2048,  128, 0, stream>>>(x, ws);
    k_dft_h   <<<1024,  128, 0, stream>>>(ws);
    k_modemix <<<8192,  128, 0, stream>>>(ws);
    k_idft_h  <<<4096,  128, 0, stream>>>(ws);
    k_idft_w  <<<16384, 128, 0, stream>>>(ws, out);
}